// PoseNeuralCDE_18262200942805
// MI455X (gfx1250) — compile-verified
//
#include <hip/hip_runtime.h>
#include <math.h>

typedef __attribute__((ext_vector_type(2))) float v2f;
typedef __attribute__((ext_vector_type(8))) float v8f;

#define B_ 512
#define F_ 768
#define H_ 1024

// ---------------------------------------------------------------------------
// activations
// ---------------------------------------------------------------------------
__device__ __forceinline__ float softplus_f(float x) {
  // jax.nn.softplus = logaddexp(x, 0) = max(x,0) + log1p(exp(-|x|))
  return fmaxf(x, 0.0f) + log1pf(expf(-fabsf(x)));
}
__device__ __forceinline__ float act_apply(float x, int act) {
  if (act == 1) return softplus_f(x);
  if (act == 2) return x > 0.0f ? x : 0.1f * x;
  return x;
}

// ---------------------------------------------------------------------------
// FP32 WMMA GEMM core.  NGUARD=false -> branch-free hot loop (all big GEMMs).
// NGUARD=true  -> out-of-range columns: address clamped to row 0, fragment
//                 zeroed with per-lane selects (no exec-mask branching).
// ---------------------------------------------------------------------------
template <bool NGUARD>
__device__ __forceinline__ void gemm_core(const float* __restrict__ A,
                                          const float* __restrict__ W,
                                          int N, int K, int arow0, int ncol0,
                                          int koff, v8f acc[2][2]) {
  const float* ap0 = A + (size_t)arow0 * K + koff;
  const float* ap1 = ap0 + (size_t)16 * K;

  const int c0 = ncol0;
  const int c1 = ncol0 + 16;
  bool v0 = true, v1 = true;
  const float* bp0;
  const float* bp1;
  if (NGUARD) {
    v0 = (c0 < N);
    v1 = (c1 < N);
    bp0 = W + (size_t)(v0 ? c0 : 0) * K + koff;
    bp1 = W + (size_t)(v1 ? c1 : 0) * K + koff;
  } else {
    bp0 = W + (size_t)c0 * K + koff;
    bp1 = W + (size_t)c1 * K + koff;
  }

#pragma unroll 4
  for (int k0 = 0; k0 < K; k0 += 4) {
    v2f a0, a1, b0, b1;
    a0.x = ap0[0]; a0.y = ap0[1];
    a1.x = ap1[0]; a1.y = ap1[1];
    b0.x = bp0[0]; b0.y = bp0[1];
    b1.x = bp1[0]; b1.y = bp1[1];
    if (NGUARD) {
      b0.x = v0 ? b0.x : 0.0f;  b0.y = v0 ? b0.y : 0.0f;
      b1.x = v1 ? b1.x : 0.0f;  b1.y = v1 ? b1.y : 0.0f;
    }
    acc[0][0] = __builtin_amdgcn_wmma_f32_16x16x4_f32(false, a0, false, b0,
                                                      (short)0, acc[0][0], false, false);
    acc[0][1] = __builtin_amdgcn_wmma_f32_16x16x4_f32(false, a0, false, b1,
                                                      (short)0, acc[0][1], false, false);
    acc[1][0] = __builtin_amdgcn_wmma_f32_16x16x4_f32(false, a1, false, b0,
                                                      (short)0, acc[1][0], false, false);
    acc[1][1] = __builtin_amdgcn_wmma_f32_16x16x4_f32(false, a1, false, b1,
                                                      (short)0, acc[1][1], false, false);
    ap0 += 4; ap1 += 4; bp0 += 4; bp1 += 4;
  }
}

// ---------------------------------------------------------------------------
// FP32 WMMA GEMM:  C[M,N] = act(A[M,K] @ W[N,K]^T + bias[N])
// M is a multiple of 64 (grid.y = M/64).  4 waves/block, 32x32 tile per wave.
//
// Fragment layouts (ISA 7.12.2, 32-bit):
//   A 16x4 : VGPR0 = K0 (lanes0-15) / K2 (lanes16-31); VGPR1 = K1 / K3
//   B 4x16 : VGPR0 = K0 row (lanes=N 0..15) / K2 row;  VGPR1 = K1 / K3
//   C 16x16: VGPR v = row v (lanes0-15) / row v+8 (lanes16-31), lane%16 = col
// ---------------------------------------------------------------------------
__global__ __launch_bounds__(128)
void gemm_wmma_f32(const float* __restrict__ A, const float* __restrict__ W,
                   const float* __restrict__ bias, float* __restrict__ C,
                   int N, int K, int act) {
  const int lane = threadIdx.x & 31;
  const int wid  = threadIdx.x >> 5;
  const int mbase = blockIdx.y * 64 + (wid >> 1) * 32;
  const int nbase = blockIdx.x * 64 + (wid & 1) * 32;
  const int l15  = lane & 15;
  const int koff = (lane < 16) ? 0 : 2;

  v8f acc[2][2] = {};

  const int arow0 = mbase + l15;
  const int ncol0 = nbase + l15;

  if (nbase + 32 <= N) {
    gemm_core<false>(A, W, N, K, arow0, ncol0, koff, acc);  // hot, branch-free
  } else {
    gemm_core<true>(A, W, N, K, arow0, ncol0, koff, acc);   // cold, guarded
  }

  const int rowadd = (lane < 16) ? 0 : 8;
#pragma unroll
  for (int j = 0; j < 2; ++j) {
    const int col = nbase + j * 16 + l15;
    if (col >= N) continue;
    const float bv = bias[col];
#pragma unroll
    for (int i = 0; i < 2; ++i) {
#pragma unroll
      for (int v = 0; v < 8; ++v) {
        const int row = mbase + i * 16 + rowadd + v;
        C[(size_t)row * N + col] = act_apply(acc[i][j][v] + bv, act);
      }
    }
  }
}

// ---------------------------------------------------------------------------
// init: fused = concat(fv, fi); y = 0; t/t1/dt/done per-row state
// ---------------------------------------------------------------------------
__global__ void init_k(const float* __restrict__ fv, const float* __restrict__ fi,
                       const float* __restrict__ ts, float* __restrict__ fused,
                       float* __restrict__ y, float* __restrict__ t,
                       float* __restrict__ t1, float* __restrict__ dt,
                       int* __restrict__ done) {
  int i = blockIdx.x * blockDim.x + threadIdx.x;
  if (i < B_ * F_) {
    int m = i / F_, f = i % F_;
    fused[i] = (f < 512) ? fv[m * 512 + f] : fi[m * 256 + (f - 512)];
    y[i] = 0.0f;
  }
  if (i < B_) {
    float a = ts[i * 2 + 0], b = ts[i * 2 + 1];
    t[i] = a; t1[i] = b; dt[i] = (b - a) * 0.01f; done[i] = 0;
  }
}

// dtc = min(dt, t1 - t)
__global__ void step_begin_k(const float* __restrict__ dt, const float* __restrict__ t,
                             const float* __restrict__ t1, float* __restrict__ dtc) {
  int i = blockIdx.x * blockDim.x + threadIdx.x;
  if (i < B_) dtc[i] = fminf(dt[i], t1[i] - t[i]);
}

// out = y + dtc[row] * (c1*p1 + c2*p2 + c3*p3 + c4*p4 + c5*p5)
__global__ void combine5_k(float* __restrict__ out, const float* __restrict__ y,
                           const float* __restrict__ dtc,
                           const float* __restrict__ p1, float c1,
                           const float* __restrict__ p2, float c2,
                           const float* __restrict__ p3, float c3,
                           const float* __restrict__ p4, float c4,
                           const float* __restrict__ p5, float c5) {
  int i = blockIdx.x * blockDim.x + threadIdx.x;
  if (i >= B_ * F_) return;
  float d = dtc[i / F_];
  float s = c1 * p1[i] + c2 * p2[i] + c3 * p3[i] + c4 * p4[i] + c5 * p5[i];
  out[i] = y[i] + d * s;
}

// ---------------------------------------------------------------------------
// per-row error norm + accept/step-size update (one block per batch row)
// ---------------------------------------------------------------------------
__global__ __launch_bounds__(256)
void ode_update_k(float* __restrict__ y, const float* __restrict__ y5,
                  const float* __restrict__ k1, const float* __restrict__ k3,
                  const float* __restrict__ k4, const float* __restrict__ k5,
                  const float* __restrict__ k6, const float* __restrict__ k7,
                  const float* __restrict__ dtc, float* __restrict__ t,
                  const float* __restrict__ t1, float* __restrict__ dt,
                  int* __restrict__ done) {
  const float e1 = (float)(71.0 / 57600.0);
  const float e3 = (float)(-71.0 / 16695.0);
  const float e4 = (float)(71.0 / 1920.0);
  const float e5 = (float)(-17253.0 / 339200.0);
  const float e6 = (float)(22.0 / 525.0);
  const float e7 = (float)(-1.0 / 40.0);

  const int m = blockIdx.x;
  const int tid = threadIdx.x;
  __shared__ float red[256];
  __shared__ float s_accept;

  const float d = dtc[m];
  float local = 0.0f;
  for (int f = tid; f < F_; f += 256) {
    size_t i = (size_t)m * F_ + f;
    float err = d * (e1 * k1[i] + e3 * k3[i] + e4 * k4[i] +
                     e5 * k5[i] + e6 * k6[i] + e7 * k7[i]);
    float yv = y[i], y5v = y5[i];
    float scale = 1e-9f + 1e-6f * fmaxf(fabsf(yv), fabsf(y5v));
    float q = err / scale;
    local += q * q;
  }
  red[tid] = local;
  __syncthreads();
  for (int s = 128; s > 0; s >>= 1) {
    if (tid < s) red[tid] += red[tid + s];
    __syncthreads();
  }
  if (tid == 0) {
    float enorm = fmaxf(sqrtf(red[0] / (float)F_), 1e-10f);
    int dn = done[m];
    int accept = (enorm <= 1.0f) && !dn;
    float tnew = accept ? (t[m] + d) : t[m];
    int done_new = (tnew >= t1[m] - 1e-12f) ? 1 : 0;
    float fac = 0.9f * powf(enorm, -0.2f);
    fac = fminf(fmaxf(fac, 0.2f), 10.0f);
    float dtn = done_new ? dt[m] : d * fac;
    t[m] = tnew; dt[m] = dtn; done[m] = done_new;
    s_accept = accept ? 1.0f : 0.0f;
  }
  __syncthreads();
  if (s_accept != 0.0f) {
    for (int f = tid; f < F_; f += 256) {
      size_t i = (size_t)m * F_ + f;
      y[i] = y5[i];
    }
  }
}

// ---------------------------------------------------------------------------
// GRU gates: writes h_new to ws and to d_out (offset 512*6)
// ---------------------------------------------------------------------------
__global__ void gru_k(const float* __restrict__ gx, const float* __restrict__ gh,
                      const float* __restrict__ yF, float* __restrict__ hnew,
                      float* __restrict__ out_hnew) {
  int i = blockIdx.x * blockDim.x + threadIdx.x;
  if (i >= B_ * F_) return;
  int m = i / F_, f = i % F_;
  size_t base = (size_t)m * (3 * F_);
  float xr = gx[base + f], xz = gx[base + F_ + f], xn = gx[base + 2 * F_ + f];
  float hr = gh[base + f], hz = gh[base + F_ + f], hn = gh[base + 2 * F_ + f];
  float r = 1.0f / (1.0f + expf(-(xr + hr)));
  float z = 1.0f / (1.0f + expf(-(xz + hz)));
  float n = tanhf(xn + r * hn);
  float h = (1.0f - z) * n + z * yF[i];
  hnew[i] = h;
  out_hnew[i] = h;
}

// ---------------------------------------------------------------------------
// host orchestration
// ---------------------------------------------------------------------------
extern "C" void kernel_launch(void* const* d_in, const int* in_sizes, int n_in,
                              void* d_out, int out_size, void* d_ws, size_t ws_size,
                              hipStream_t stream) {
  (void)in_sizes; (void)n_in; (void)out_size; (void)ws_size;

  const float* fv   = (const float*)d_in[0];
  const float* fi   = (const float*)d_in[2];
  const float* ts   = (const float*)d_in[4];
  const float* W0   = (const float*)d_in[5];   // [1024,768]
  const float* W1   = (const float*)d_in[6];   // [1024,1024]
  const float* W2   = (const float*)d_in[7];   // [768,1024]
  const float* b0   = (const float*)d_in[8];
  const float* b1   = (const float*)d_in[9];
  const float* b2   = (const float*)d_in[10];
  const float* gWih = (const float*)d_in[11];  // [2304,768]
  const float* gWhh = (const float*)d_in[12];  // [2304,768]
  const float* gbih = (const float*)d_in[13];
  const float* gbhh = (const float*)d_in[14];
  const float* rW0  = (const float*)d_in[15];  // [128,768]
  const float* rW1  = (const float*)d_in[16];  // [256,128]
  const float* rW2  = (const float*)d_in[17];  // [128,256]
  const float* rW3  = (const float*)d_in[18];  // [6,128]
  const float* rb0  = (const float*)d_in[19];
  const float* rb1  = (const float*)d_in[20];
  const float* rb2  = (const float*)d_in[21];
  const float* rb3  = (const float*)d_in[22];

  float* out = (float*)d_out;          // [512*6] pose, then [512*768] h_new

  // workspace carve-out
  float* ws = (float*)d_ws;
  size_t off = 0;
  auto carve = [&](size_t n) { float* p = ws + off; off += n; return p; };
  float* fused = carve((size_t)B_ * F_);
  float* y     = carve((size_t)B_ * F_);
  float* ytmp  = carve((size_t)B_ * F_);
  float* y5    = carve((size_t)B_ * F_);
  float* h1    = carve((size_t)B_ * H_);
  float* h2    = carve((size_t)B_ * H_);
  float* kk[8];
  for (int i = 1; i <= 7; ++i) kk[i] = carve((size_t)B_ * F_);
  float* gx    = carve((size_t)B_ * 3 * F_);
  float* gh    = carve((size_t)B_ * 3 * F_);
  float* hnew  = carve((size_t)B_ * F_);
  float* r1    = carve((size_t)B_ * 128);
  float* r2    = carve((size_t)B_ * 256);
  float* r3    = carve((size_t)B_ * 128);
  float* tv    = carve(B_);
  float* t1v   = carve(B_);
  float* dtv   = carve(B_);
  float* dtcv  = carve(B_);
  int*   done  = (int*)carve(B_);

  const int NEL = B_ * F_;
  const dim3 eb(256);
  const dim3 eg((NEL + 255) / 256);

  auto gemm = [&](const float* A, const float* Wm, const float* bias, float* C,
                  int N, int K, int act) {
    dim3 g((N + 63) / 64, B_ / 64);
    gemm_wmma_f32<<<g, dim3(128), 0, stream>>>(A, Wm, bias, C, N, K, act);
  };
  auto mlp = [&](const float* in, float* outk) {
    gemm(in, W0, b0, h1, H_, F_, 1);   // softplus
    gemm(h1, W1, b1, h2, H_, H_, 1);   // softplus
    gemm(h2, W2, b2, outk, F_, H_, 0); // linear
  };
  auto combine = [&](float* o, const float* p1, float c1, const float* p2, float c2,
                     const float* p3, float c3, const float* p4, float c4,
                     const float* p5, float c5) {
    combine5_k<<<eg, eb, 0, stream>>>(o, y, dtcv, p1, c1, p2, c2, p3, c3, p4, c4, p5, c5);
  };

  init_k<<<eg, eb, 0, stream>>>(fv, fi, ts, fused, y, tv, t1v, dtv, done);

  // Dormand-Prince RK45 coefficients
  const float a21 = 0.2f;
  const float a31 = (float)(3.0 / 40.0),     a32 = (float)(9.0 / 40.0);
  const float a41 = (float)(44.0 / 45.0),    a42 = (float)(-56.0 / 15.0),   a43 = (float)(32.0 / 9.0);
  const float a51 = (float)(19372.0/6561.0), a52 = (float)(-25360.0/2187.0),
              a53 = (float)(64448.0/6561.0), a54 = (float)(-212.0/729.0);
  const float a61 = (float)(9017.0/3168.0),  a62 = (float)(-355.0/33.0),
              a63 = (float)(46732.0/5247.0), a64 = (float)(49.0/176.0), a65 = (float)(-5103.0/18656.0);
  const float bb1 = (float)(35.0/384.0),     bb3 = (float)(500.0/1113.0),
              bb4 = (float)(125.0/192.0),    bb5 = (float)(-2187.0/6784.0), bb6 = (float)(11.0/84.0);

  for (int s = 0; s < 32; ++s) {
    step_begin_k<<<dim3(2), eb, 0, stream>>>(dtv, tv, t1v, dtcv);
    mlp(y, kk[1]);
    combine(ytmp, kk[1], a21, kk[1], 0.f, kk[1], 0.f, kk[1], 0.f, kk[1], 0.f);
    mlp(ytmp, kk[2]);
    combine(ytmp, kk[1], a31, kk[2], a32, kk[1], 0.f, kk[1], 0.f, kk[1], 0.f);
    mlp(ytmp, kk[3]);
    combine(ytmp, kk[1], a41, kk[2], a42, kk[3], a43, kk[1], 0.f, kk[1], 0.f);
    mlp(ytmp, kk[4]);
    combine(ytmp, kk[1], a51, kk[2], a52, kk[3], a53, kk[4], a54, kk[1], 0.f);
    mlp(ytmp, kk[5]);
    combine(ytmp, kk[1], a61, kk[2], a62, kk[3], a63, kk[4], a64, kk[5], a65);
    mlp(ytmp, kk[6]);
    combine(y5, kk[1], bb1, kk[3], bb3, kk[4], bb4, kk[5], bb5, kk[6], bb6);
    mlp(y5, kk[7]);
    ode_update_k<<<dim3(B_), eb, 0, stream>>>(y, y5, kk[1], kk[3], kk[4], kk[5],
                                              kk[6], kk[7], dtcv, tv, t1v, dtv, done);
  }

  // GRU cell
  gemm(fused, gWih, gbih, gx, 3 * F_, F_, 0);
  gemm(y,     gWhh, gbhh, gh, 3 * F_, F_, 0);
  gru_k<<<eg, eb, 0, stream>>>(gx, gh, y, hnew, out + (size_t)B_ * 6);

  // regressor head
  gemm(hnew, rW0, rb0, r1, 128, F_, 2);
  gemm(r1,   rW1, rb1, r2, 256, 128, 2);
  gemm(r2,   rW2, rb2, r3, 128, 256, 2);
  gemm(r3,   rW3, rb3, out, 6, 128, 0);  // pose -> d_out[0 .. 512*6)
}